// Laplace_85538568667656
// MI455X (gfx1250) — compile-verified
//
#include <hip/hip_runtime.h>

// Laplace diagonal linear recurrence, out (T, N_S, F) f32.
// T=2048, F=256, N_S=108. Memory-bound: 226.5MB of stores -> ~9.7us floor.
// 3-phase segmented scan; phase C runs the intra-chunk triangular matmul on
// the matrix pipe with V_WMMA_F32_16X16X4_F32 (exact f32).

#define T_LEN   2048
#define F_DIM   256
#define NS      108
#define KPAR    4
#define SEG     8
#define SEG_LEN (T_LEN / SEG)          // 256
#define CHUNK   16
#define CHUNKS_PER_SEG (SEG_LEN / CHUNK) // 16

typedef __attribute__((ext_vector_type(2))) float v2f;
typedef __attribute__((ext_vector_type(8))) float v8f;

__device__ __forceinline__ void laplace_consts(int i, float& a, float& d, float& s) {
    // c = 20^(1/99)-1 ; tau*_i = (1+c)^(i-K) ; s = K/tau* ; a = exp(-s); d = (1-a)/s
    const float log1pc = logf(20.0f) * (1.0f / 99.0f);   // constant-folded
    float tau = expf((float)(i - KPAR) * log1pc);
    s = (float)KPAR / tau;
    a = expf(-s);
    d = (1.0f - a) / s;
}

// ---- Phase A: per-(i,f,segment) zero-init local end state -------------------
__global__ void lap_phaseA(const float* __restrict__ inp, float* __restrict__ E) {
    const int i   = blockIdx.x;   // channel
    const int seg = blockIdx.y;   // segment
    const int f   = threadIdx.x;  // feature
    float a, d, s; laplace_consts(i, a, d, s);
    const float* x = inp + (size_t)seg * SEG_LEN * F_DIM + f;
    float t = 0.0f;
    for (int n = 0; n < SEG_LEN; ++n)
        t = fmaf(a, t, d * x[(size_t)n * F_DIM]);
    E[((size_t)seg * NS + i) * F_DIM + f] = t;
}

// ---- Phase B: scan the 8 segment states into carry-ins ----------------------
__global__ void lap_phaseB(const float* __restrict__ E, float* __restrict__ C) {
    const int i = blockIdx.x;
    const int f = threadIdx.x;
    float a, d, s; laplace_consts(i, a, d, s);
    const float aP = expf(-s * (float)SEG_LEN);   // a^SEG_LEN
    float c = 0.0f;
    for (int seg = 0; seg < SEG; ++seg) {
        const size_t idx = ((size_t)seg * NS + i) * F_DIM + f;
        C[idx] = c;                       // carry-in for this segment
        c = fmaf(aP, c, E[idx]);
    }
}

// ---- Phase C: WMMA chunked scan, writes all outputs -------------------------
// One wave owns (channel i, 16-wide f tile, segment): 16 chunks of 16 steps.
// Per chunk: D(16x16) = Wc*carry + sum_k W[:,4k:4k+4] @ X[4k:4k+4, ftile]
__global__ __launch_bounds__(256)
void lap_phaseC(const float* __restrict__ inp, const float* __restrict__ C,
                float* __restrict__ out) {
    const int lane = threadIdx.x & 31;
    const int w    = blockIdx.x * 8 + (threadIdx.x >> 5);
    const int seg  = w & 7;
    const int ft   = (w >> 3) & 15;
    const int i    = w >> 7;

    float a, d, s; laplace_consts(i, a, d, s);

    const int  m  = lane & 15;
    const bool hi = lane >= 16;

    // A blocks (16x4 each): VGPR0 = K {0,2}, VGPR1 = K {1,3}; W[m,j]=d*a^(m-j), j<=m
    v2f A[4];
#pragma unroll
    for (int k = 0; k < 4; ++k) {
        const int j0 = 4 * k + (hi ? 2 : 0);
        const int j1 = j0 + 1;
        A[k].x = (j0 <= m) ? d * expf(-s * (float)(m - j0)) : 0.0f;
        A[k].y = (j1 <= m) ? d * expf(-s * (float)(m - j1)) : 0.0f;
    }
    // carry A block: column 0 = a^(r+1), columns 1..3 = 0
    v2f Ac;
    Ac.x = hi ? 0.0f : expf(-s * (float)(m + 1));
    Ac.y = 0.0f;

    const int fcol = ft * 16 + m;
    float carry = C[((size_t)seg * NS + i) * F_DIM + fcol];   // lo lanes used

    const float* xbase = inp + (size_t)seg * SEG_LEN * F_DIM + fcol;
    float*       obase = out + ((size_t)seg * SEG_LEN * NS + i) * F_DIM + fcol;

    for (int ch = 0; ch < CHUNKS_PER_SEG; ++ch) {
        const float* xc = xbase + (size_t)ch * CHUNK * F_DIM;

        // B blocks (4x16): VGPR0 = rows {4k+0, 4k+2}, VGPR1 = rows {4k+1, 4k+3}
        v2f B[4];
#pragma unroll
        for (int k = 0; k < 4; ++k) {
            const int r0 = 4 * k + (hi ? 2 : 0);
            B[k].x = xc[(size_t)r0 * F_DIM];
            B[k].y = xc[(size_t)(r0 + 1) * F_DIM];
        }
        v2f Bc; Bc.x = hi ? 0.0f : carry; Bc.y = 0.0f;

        v8f acc = {};
        acc = __builtin_amdgcn_wmma_f32_16x16x4_f32(false, Ac, false, Bc,
                                                    (short)0, acc, false, false);
#pragma unroll
        for (int k = 0; k < 4; ++k)
            acc = __builtin_amdgcn_wmma_f32_16x16x4_f32(false, A[k], false, B[k],
                                                        (short)0, acc, false, false);

        // D layout: VGPR r = rows {r (lo lanes), r+8 (hi lanes)}, N = lane%16.
        // Output is write-once streaming (226MB > L2): non-temporal stores.
        float* oc = obase + (size_t)ch * CHUNK * NS * F_DIM
                          + (hi ? (size_t)8 * NS * F_DIM : (size_t)0);
#pragma unroll
        for (int r = 0; r < 8; ++r)
            __builtin_nontemporal_store(acc[r], oc + (size_t)r * NS * F_DIM);

        // next carry = row 15 = acc[7] in hi lanes -> broadcast to lo lanes
        carry = __shfl(acc[7], 16 + m, 32);
    }
}

extern "C" void kernel_launch(void* const* d_in, const int* in_sizes, int n_in,
                              void* d_out, int out_size, void* d_ws, size_t ws_size,
                              hipStream_t stream) {
    (void)in_sizes; (void)n_in; (void)out_size; (void)ws_size;
    const float* inp = (const float*)d_in[0];
    float*       out = (float*)d_out;
    float* E = (float*)d_ws;                          // SEG*NS*F floats
    float* C = E + (size_t)SEG * NS * F_DIM;          // SEG*NS*F floats

    lap_phaseA<<<dim3(NS, SEG), F_DIM, 0, stream>>>(inp, E);
    lap_phaseB<<<NS, F_DIM, 0, stream>>>(E, C);
    lap_phaseC<<<(NS * (F_DIM / 16) * SEG) / 8, 256, 0, stream>>>(inp, C, out);
}